// GymNetwork_21208548507777
// MI455X (gfx1250) — compile-verified
//
#include <hip/hip_runtime.h>

typedef float v2f __attribute__((ext_vector_type(2)));
typedef float v8f __attribute__((ext_vector_type(8)));

#define BATCH   262144
#define DIM     128
#define NGAMES  8
#define FEAT    80
#define NACT    18

#define TILE_M          16
#define WAVES_PER_BLOCK 4

// V_WMMA_F32_16X16X4_F32 fragment mapping (wave32):
//   A frag: (M = lane&15, K = 2*(lane>>4) + {0,1})  -> one b64 load, K-contig
//   B frag: (N = lane&15, K = 2*(lane>>4) + {0,1})  -> one b64 load, K-contig
//   C frag: element i at (M = i + 8*(lane>>4), N = lane&15)

__global__ __launch_bounds__(32 * WAVES_PER_BLOCK)
void gym_mlp_wmma_kernel(const float* __restrict__ state,  // [B, D]
                         const int*   __restrict__ idx,    // [B], sorted
                         const float* __restrict__ W1,     // [G, F, D]
                         const float* __restrict__ b1,     // [G, F]
                         const float* __restrict__ W2,     // [F, F]
                         const float* __restrict__ b2,     // [F]
                         const float* __restrict__ W3,     // [F, F]
                         const float* __restrict__ b3,     // [F]
                         const float* __restrict__ W4,     // [G, A, F]
                         const float* __restrict__ b4,     // [G, A]
                         float*       __restrict__ out)    // [B, A]
{
  __shared__ __align__(16) float lds_h[WAVES_PER_BLOCK][2][TILE_M * FEAT];

  const int lane = threadIdx.x & 31;
  const int wib  = threadIdx.x >> 5;
  const int ln   = lane & 15;  // N (B/C frags) or M (A frags)
  const int lh   = lane >> 4;  // K-pair selector / C M-half selector
  const int tile = blockIdx.x * WAVES_PER_BLOCK + wib;
  const int row0 = tile * TILE_M;

  float* hA = &lds_h[wib][0][0];
  float* hB = &lds_h[wib][1][0];

  // Per-lane game ids of the 8 C-fragment rows this lane owns (M = i + 8*lh).
  int rIdx[8];
#pragma unroll
  for (int i = 0; i < 8; ++i) rIdx[i] = idx[row0 + i + 8 * lh];

  // idx is sorted -> tile spans games [gfirst, glast] (almost always 1).
  // Force scalar so the routing loop is SALU-driven and WMMAs see full EXEC.
  const int gfirst = __builtin_amdgcn_readfirstlane(idx[row0]);
  const int glast  = __builtin_amdgcn_readfirstlane(idx[row0 + TILE_M - 1]);

  // Nudge shared weights toward cache (global_prefetch_b8).
  __builtin_prefetch(W2 + lane * 16, 0, 0);
  __builtin_prefetch(W3 + lane * 16, 0, 0);

  // ---- Load this tile's A fragments of state ONCE (K = 0..127) -------------
  v2f aS[DIM / 4];
  {
    const float* arow = state + (size_t)(row0 + ln) * DIM + 2 * lh;
#pragma unroll
    for (int kb = 0; kb < DIM; kb += 4) aS[kb / 4] = *(const v2f*)(arow + kb);
  }

  // ---------- Layer 1 (routed): h1 = relu(state @ W1[g]^T + b1[g]) ----------
  for (int g = gfirst; g <= glast; ++g) {               // scalar-uniform bounds
    const float* W1g = W1 + (size_t)g * FEAT * DIM;
    const float* b1g = b1 + g * FEAT;
#pragma unroll
    for (int nt = 0; nt < FEAT / 16; ++nt) {
      const int n0 = nt * 16;
      v8f c = {0.f, 0.f, 0.f, 0.f, 0.f, 0.f, 0.f, 0.f};
      const float* brow = W1g + (size_t)(n0 + ln) * DIM + 2 * lh;
#pragma unroll
      for (int kb = 0; kb < DIM; kb += 4) {
        v2f b = *(const v2f*)(brow + kb);
        c = __builtin_amdgcn_wmma_f32_16x16x4_f32(false, aS[kb / 4], false, b,
                                                  (short)0, c, false, false);
      }
      const float bv = b1g[n0 + ln];
#pragma unroll
      for (int i = 0; i < 8; ++i) {
        float v = c[i] + bv;
        v = v > 0.f ? v : 0.f;
        if (rIdx[i] == g)                      // merge only this game's rows
          hA[(i + 8 * lh) * FEAT + n0 + ln] = v;
      }
    }
  }
  __syncthreads();

  // ---------- Layers 2-4 share the same A-row hoist pattern -----------------
  v2f aH[FEAT / 4];

  // Layer 2: hB = relu(hA @ W2^T + b2)
#pragma unroll
  for (int kb = 0; kb < FEAT; kb += 4)
    aH[kb / 4] = *(const v2f*)(hA + ln * FEAT + 2 * lh + kb);
#pragma unroll
  for (int nt = 0; nt < FEAT / 16; ++nt) {
    const int n0 = nt * 16;
    v8f c = {0.f, 0.f, 0.f, 0.f, 0.f, 0.f, 0.f, 0.f};
    const float* brow = W2 + (size_t)(n0 + ln) * FEAT + 2 * lh;
#pragma unroll
    for (int kb = 0; kb < FEAT; kb += 4) {
      v2f b = *(const v2f*)(brow + kb);
      c = __builtin_amdgcn_wmma_f32_16x16x4_f32(false, aH[kb / 4], false, b,
                                                (short)0, c, false, false);
    }
    const float bv = b2[n0 + ln];
#pragma unroll
    for (int i = 0; i < 8; ++i) {
      float v = c[i] + bv;
      hB[(i + 8 * lh) * FEAT + n0 + ln] = v > 0.f ? v : 0.f;
    }
  }
  __syncthreads();

  // Layer 3: hA = relu(hB @ W3^T + b3)
#pragma unroll
  for (int kb = 0; kb < FEAT; kb += 4)
    aH[kb / 4] = *(const v2f*)(hB + ln * FEAT + 2 * lh + kb);
#pragma unroll
  for (int nt = 0; nt < FEAT / 16; ++nt) {
    const int n0 = nt * 16;
    v8f c = {0.f, 0.f, 0.f, 0.f, 0.f, 0.f, 0.f, 0.f};
    const float* brow = W3 + (size_t)(n0 + ln) * FEAT + 2 * lh;
#pragma unroll
    for (int kb = 0; kb < FEAT; kb += 4) {
      v2f b = *(const v2f*)(brow + kb);
      c = __builtin_amdgcn_wmma_f32_16x16x4_f32(false, aH[kb / 4], false, b,
                                                (short)0, c, false, false);
    }
    const float bv = b3[n0 + ln];
#pragma unroll
    for (int i = 0; i < 8; ++i) {
      float v = c[i] + bv;
      hA[(i + 8 * lh) * FEAT + n0 + ln] = v > 0.f ? v : 0.f;
    }
  }
  __syncthreads();

  // ---------- Layer 4 (routed head): q = h @ W4[g]^T + b4[g], A=18 ----------
#pragma unroll
  for (int kb = 0; kb < FEAT; kb += 4)
    aH[kb / 4] = *(const v2f*)(hA + ln * FEAT + 2 * lh + kb);

  for (int g = gfirst; g <= glast; ++g) {               // scalar-uniform bounds
    const float* W4g = W4 + (size_t)g * NACT * FEAT;
    const float* b4g = b4 + g * NACT;
#pragma unroll
    for (int nt = 0; nt < 2; ++nt) {                    // 18 cols -> 2 tiles
      const int n0 = nt * 16;
      const int n  = n0 + ln;
      const bool nvalid = (n < NACT);
      v8f c = {0.f, 0.f, 0.f, 0.f, 0.f, 0.f, 0.f, 0.f};
      const float* brow = W4g + (size_t)n * FEAT + 2 * lh;
#pragma unroll
      for (int kb = 0; kb < FEAT; kb += 4) {
        v2f b = {0.f, 0.f};
        if (nvalid) b = *(const v2f*)(brow + kb);       // mask padded columns
        c = __builtin_amdgcn_wmma_f32_16x16x4_f32(false, aH[kb / 4], false, b,
                                                  (short)0, c, false, false);
      }
#pragma unroll
      for (int i = 0; i < 8; ++i) {
        if (nvalid && rIdx[i] == g)
          out[(size_t)(row0 + i + 8 * lh) * NACT + n] = c[i] + b4g[n];
      }
    }
  }
}

extern "C" void kernel_launch(void* const* d_in, const int* in_sizes, int n_in,
                              void* d_out, int out_size, void* d_ws, size_t ws_size,
                              hipStream_t stream) {
  (void)in_sizes; (void)n_in; (void)out_size; (void)d_ws; (void)ws_size;
  const float* state = (const float*)d_in[0];
  const int*   idx   = (const int*)  d_in[1];
  const float* W1    = (const float*)d_in[2];
  const float* b1    = (const float*)d_in[3];
  const float* W2    = (const float*)d_in[4];
  const float* b2    = (const float*)d_in[5];
  const float* W3    = (const float*)d_in[6];
  const float* b3    = (const float*)d_in[7];
  const float* W4    = (const float*)d_in[8];
  const float* b4    = (const float*)d_in[9];
  float* out = (float*)d_out;

  const int tiles  = BATCH / TILE_M;                 // 16384 waves
  const int blocks = tiles / WAVES_PER_BLOCK;        // 4096 blocks of 128 thr
  gym_mlp_wmma_kernel<<<blocks, 32 * WAVES_PER_BLOCK, 0, stream>>>(
      state, idx, W1, b1, W2, b2, W3, b3, W4, b4, out);
}